// PolyConvFrame_86612310491927
// MI455X (gfx1250) — compile-verified
//
#include <hip/hip_runtime.h>
#include <math.h>

typedef float v4f __attribute__((ext_vector_type(4)));

// Jacobi recurrence coefficients for alpha=1.0, beta=0.2, DEPTH=3,
// computed in double precision, rounded once to fp32.
//   x1 = 0.4 + 1.6*x
//   x2 = (A2*x + B2)*x1 - C2           A2=1.70625       B2=0.0984375      C2=0.609375
//   x3 = (A3*x + B3)*x2 - C3*x1        A3=1.77142857    B3=0.0454212454   C3=0.7252747253
__device__ __forceinline__ float jacobi_eval(float x, float t0, float t1,
                                             float t2, float t3) {
    const float x1 = fmaf(1.6f, x, 0.4f);
    const float x2 = fmaf(fmaf(1.70625f, x, 0.0984375f), x1, -0.609375f);
    const float x3 = fmaf(fmaf(1.7714285714285714f, x, 0.045421245421245421f),
                          x2, -0.72527472527472525f * x1);
    float r = fmaf(t1, x1, t0);
    r = fmaf(t2, x2, r);
    r = fmaf(t3, x3, r);
    return r;
}

__device__ __forceinline__ v4f jacobi_eval4(v4f v, float t0, float t1,
                                            float t2, float t3) {
    v4f r;
    r.x = jacobi_eval(v.x, t0, t1, t2, t3);
    r.y = jacobi_eval(v.y, t0, t1, t2, t3);
    r.z = jacobi_eval(v.z, t0, t1, t2, t3);
    r.w = jacobi_eval(v.w, t0, t1, t2, t3);
    return r;
}

// One-thread setup: t[L] = tanh(theta_L) into workspace (scalar-float pipe),
// plus the scalar tail (n % 4) so the streaming kernels stay pure b128.
__global__ void poly_setup(const float* __restrict__ thetas,
                           const float* __restrict__ adj,
                           float* __restrict__ out,
                           float* __restrict__ tws,
                           long long n) {
    if (threadIdx.x == 0 && blockIdx.x == 0) {
        const float t0 = tanhf(thetas[0]);
        const float t1 = tanhf(thetas[1]);
        const float t2 = tanhf(thetas[2]);
        const float t3 = tanhf(thetas[3]);
        tws[0] = t0; tws[1] = t1; tws[2] = t2; tws[3] = t3;
        const long long tail = (n / 4) * 4;
        for (long long i = tail; i < n; ++i) {
            out[i] = jacobi_eval(adj[i], t0, t1, t2, t3);
        }
    }
}

#define VEC_PER_THREAD 4   // 4 x 16 B = 64 B in flight per thread

// Branchless bulk kernel: grid covers exactly gridDim.x * 1024 float4s.
// 4 NT b128 loads -> 128 FMAs + 16 muls -> 4 NT b128 stores per thread, no compares.
__global__ __launch_bounds__(256) void poly_stream_exact(
    const float* __restrict__ adj, float* __restrict__ out,
    const float* __restrict__ tws) {
    const v4f* __restrict__ in4 = reinterpret_cast<const v4f*>(adj);
    v4f* __restrict__ out4 = reinterpret_cast<v4f*>(out);

    const v4f t = *reinterpret_cast<const v4f*>(tws);   // one uniform b128 load

    const long long base =
        (long long)blockIdx.x * (256LL * VEC_PER_THREAD) + threadIdx.x;

    v4f v[VEC_PER_THREAD];
#pragma unroll
    for (int j = 0; j < VEC_PER_THREAD; ++j) {
        v[j] = __builtin_nontemporal_load(&in4[base + (long long)j * 256]);
    }
#pragma unroll
    for (int j = 0; j < VEC_PER_THREAD; ++j) {
        const v4f r = jacobi_eval4(v[j], t.x, t.y, t.z, t.w);
        __builtin_nontemporal_store(r, &out4[base + (long long)j * 256]);
    }
}

// Predicated kernel for the leftover float4 range [start, n4).
__global__ __launch_bounds__(256) void poly_stream_guard(
    const float* __restrict__ adj, float* __restrict__ out,
    const float* __restrict__ tws, long long start, long long n4) {
    const v4f* __restrict__ in4 = reinterpret_cast<const v4f*>(adj);
    v4f* __restrict__ out4 = reinterpret_cast<v4f*>(out);

    const v4f t = *reinterpret_cast<const v4f*>(tws);

    const long long i =
        start + (long long)blockIdx.x * 256 + threadIdx.x;
    if (i < n4) {
        const v4f v = __builtin_nontemporal_load(&in4[i]);
        const v4f r = jacobi_eval4(v, t.x, t.y, t.z, t.w);
        __builtin_nontemporal_store(r, &out4[i]);
    }
}

extern "C" void kernel_launch(void* const* d_in, const int* in_sizes, int n_in,
                              void* d_out, int out_size, void* d_ws, size_t ws_size,
                              hipStream_t stream) {
    const float* adj    = (const float*)d_in[0];   // 8192*8192 fp32
    const float* thetas = (const float*)d_in[1];   // DEPTH+1 = 4 fp32
    float* out = (float*)d_out;
    float* tws = (float*)d_ws;                     // 4 fp32 coefficients

    const long long n  = (long long)in_sizes[0];
    const long long n4 = n / 4;

    // 1) tanh(thetas) -> d_ws (scalar-float pipe), plus scalar tail.
    poly_setup<<<1, 32, 0, stream>>>(thetas, adj, out, tws, n);

    // 2) Branchless bulk: exactly-divisible portion (entire grid for 8192^2).
    const long long perBlock   = 256LL * VEC_PER_THREAD;      // float4s per block
    const long long fullBlocks = n4 / perBlock;
    const long long nFull      = fullBlocks * perBlock;
    if (fullBlocks > 0) {
        poly_stream_exact<<<(int)fullBlocks, 256, 0, stream>>>(adj, out, tws);
    }

    // 3) Remainder float4s (empty for 8192^2).
    const long long rem = n4 - nFull;
    if (rem > 0) {
        const int remBlocks = (int)((rem + 255) / 256);
        poly_stream_guard<<<remBlocks, 256, 0, stream>>>(adj, out, tws, nFull, n4);
    }
}